// BevFormerViewTransformer_83021717831989
// MI455X (gfx1250) — compile-verified
//
#include <hip/hip_runtime.h>

typedef __attribute__((ext_vector_type(2))) float v2f;
typedef __attribute__((ext_vector_type(8))) float v8f;

#define BEV_W 256
#define BEV_H 256
#define QTOT (BEV_H * BEV_W)
#define MAXLEN (256 * 256)
#define IMW_F 928.0f
#define IMH_F 512.0f
#define EPS_F 1e-5f

// ---------------------------------------------------------------------------
// K0: fill default values. [0, o_mid) = -1.5 (query grid + restore grid),
//     [o_mid, o_end) = 0 (reference_points_rebatch). Counts written fully later.
// ---------------------------------------------------------------------------
__global__ void fill_kernel(float* __restrict__ out, long o_mid, long o_end) {
    long i = (long)blockIdx.x * blockDim.x + threadIdx.x;
    if (i < o_end) out[i] = (i < o_mid) ? -1.5f : 0.0f;
}

// ---------------------------------------------------------------------------
// K1: projection + visibility via V_WMMA_F32_16X16X4_F32.
// One WMMA per wave: A packs TWO camera matrices (cam0 -> rows 0..3,
// cam1 -> rows 8..11), B packs 16 homogeneous points (4 pillars x 4 z levels).
// D layout (16x16 f32): lane n<16, VGPR r = D[r][n]; lane n>=16, VGPR r =
// D[r+8][n-16]. So lane n holds (cx,cy,cz,cw) of point (n&15) under camera
// (n<16 ? c0 : c1) in Dm[0..3].
// grid: (QTOT/32, 3*B), block 256 (8 waves, 4 pillars per wave).
// ---------------------------------------------------------------------------
__global__ void proj_vis_wmma(const float* __restrict__ ego, int* __restrict__ vis) {
    const int tid  = threadIdx.x;
    const int lane = tid & 31;
    const int wv   = tid >> 5;
    const int bcp  = blockIdx.y;
    const int b    = bcp / 3;
    const int cp   = bcp % 3;
    const int c0   = 2 * cp;
    const int c1   = c0 + 1;
    const int hi   = lane >> 4;    // half of the wave
    const int l    = lane & 15;
    const int qbase = blockIdx.x * 32 + wv * 4;

    // --- A operand (16x4 f32: lanes 0-15 hold cols 0,1; lanes 16-31 cols 2,3)
    float a0 = 0.0f, a1 = 0.0f;
    const int col = hi ? 2 : 0;
    if (l < 4) {
        const float* Mr = ego + ((b * 6 + c0) * 4 + l) * 4;
        a0 = Mr[col]; a1 = Mr[col + 1];
    } else if (l >= 8 && l < 12) {
        const float* Mr = ego + ((b * 6 + c1) * 4 + (l - 8)) * 4;
        a0 = Mr[col]; a1 = Mr[col + 1];
    }

    // --- B operand (4x16 f32): column l = point l. Lanes 0-15 carry K rows
    // 0,1 (x,y); lanes 16-31 carry K rows 2,3 (z,1).
    const int q = qbase + (l >> 2);
    const int d = l & 3;
    const float px = ((q & 255) + 0.5f) * 0.4f - 51.2f;
    const float py = ((q >> 8) + 0.5f) * 0.4f - 51.2f;
    const float pz = 0.5f + d * (7.0f / 3.0f) - 5.0f;
    const float b0 = hi ? pz   : px;
    const float b1 = hi ? 1.0f : py;

    v2f Am = {a0, a1};
    v2f Bm = {b0, b1};
    v8f Cm = {};
    v8f Dm = __builtin_amdgcn_wmma_f32_16x16x4_f32(
        false, Am, false, Bm, (short)0, Cm, false, false);

    const float cx = Dm[0], cy = Dm[1], cz = Dm[2];
    const float zc = fmaxf(cz, EPS_F);
    const float u  = cx / zc * (1.0f / IMW_F);
    const float v  = cy / zc * (1.0f / IMH_F);
    const bool visb = (cz > EPS_F) & (v > 0.0f) & (v < 1.0f) & (u > 0.0f) & (u < 1.0f);

    const unsigned m = __builtin_amdgcn_ballot_w32(visb);
    const unsigned g = (m >> (lane & ~3)) & 0xFu;   // OR over the 4 z-levels
    if ((lane & 3) == 0) {
        const int cam = hi ? c1 : c0;
        vis[(b * 6 + cam) * QTOT + q] = (g != 0u) ? 1 : 0;
    }
}

// ---------------------------------------------------------------------------
// K2: inclusive prefix sum of visibility per (b,cam) row (Q = 65536).
// One 1024-thread block per row; 64 chunks, Hillis-Steele scan in LDS.
// ---------------------------------------------------------------------------
__global__ void scan_kernel(const int* __restrict__ vis, int* __restrict__ incl) {
    const int bc = blockIdx.x;
    __shared__ int tmp[1024];
    __shared__ int carry;
    if (threadIdx.x == 0) carry = 0;
    __syncthreads();
    const int* vrow = vis + (long)bc * QTOT;
    int* orow = incl + (long)bc * QTOT;
    for (int chunk = 0; chunk < QTOT; chunk += 1024) {
        tmp[threadIdx.x] = vrow[chunk + threadIdx.x];
        __syncthreads();
        for (int off = 1; off < 1024; off <<= 1) {
            int y = (threadIdx.x >= off) ? tmp[threadIdx.x - off] : 0;
            __syncthreads();
            tmp[threadIdx.x] += y;
            __syncthreads();
        }
        int s = tmp[threadIdx.x] + carry;
        orow[chunk + threadIdx.x] = s;
        __syncthreads();
        if (threadIdx.x == 1023) carry = s;
        __syncthreads();
    }
}

// ---------------------------------------------------------------------------
// K3: scatter packed query grid (out1) and packed reference points (out3).
// Projection recomputed scalarly (12 FMAs / z-level, uniform matrix per block).
// grid: (QTOT/256, 6*B)
// ---------------------------------------------------------------------------
__global__ void scatter_q(const float* __restrict__ ego, const int* __restrict__ vis,
                          const int* __restrict__ incl,
                          float* __restrict__ out1, float* __restrict__ out3) {
    const int bc = blockIdx.y;
    const int q  = blockIdx.x * blockDim.x + threadIdx.x;
    if (q >= QTOT) return;
    if (!vis[(long)bc * QTOT + q]) return;
    const int rank = incl[(long)bc * QTOT + q] - 1;
    const int qx = q & 255, qy = q >> 8;

    const long slot = (long)bc * MAXLEN + rank;
    out1[slot * 2 + 0] = qx * (2.0f / 255.0f) - 1.0f;
    out1[slot * 2 + 1] = qy * (2.0f / 255.0f) - 1.0f;

    const float* M = ego + bc * 16;
    const float px = (qx + 0.5f) * 0.4f - 51.2f;
    const float py = (qy + 0.5f) * 0.4f - 51.2f;
    for (int d = 0; d < 4; ++d) {
        const float pz = 0.5f + d * (7.0f / 3.0f) - 5.0f;
        const float cx = M[0] * px + M[1] * py + M[2]  * pz + M[3];
        const float cy = M[4] * px + M[5] * py + M[6]  * pz + M[7];
        const float cz = M[8] * px + M[9] * py + M[10] * pz + M[11];
        const float zc = fmaxf(cz, EPS_F);
        float u = cx / zc * (1.0f / IMW_F);
        float v = cy / zc * (1.0f / IMH_F);
        u = fminf(fmaxf(u, -2.1f), 2.1f);
        v = fminf(fmaxf(v, -2.1f), 2.1f);
        const long base = (slot * 4 + d) * 2;
        out3[base + 0] = u;
        out3[base + 1] = v;
    }
}

// ---------------------------------------------------------------------------
// K4: restore_bev_grid (out2) + bev_pillar_counts (out4).
// Each (b,q) thread walks the 6 cameras; the j-th visible camera writes row
// block j ("over"). grid: (QTOT/256, B)
// ---------------------------------------------------------------------------
__global__ void restore_counts(const int* __restrict__ vis, const int* __restrict__ incl,
                               float* __restrict__ out2, float* __restrict__ out4) {
    const int b = blockIdx.y;
    const int q = blockIdx.x * blockDim.x + threadIdx.x;
    if (q >= QTOT) return;
    const int qx = q & 255, qy = q >> 8;
    int count = 0, over = 0;
    for (int c = 0; c < 6; ++c) {
        if (vis[((long)(b * 6 + c)) * QTOT + q]) {
            const int rank = incl[((long)(b * 6 + c)) * QTOT + q] - 1;
            const float gx = (rank & 255) * (2.0f / 255.0f) - 1.0f;
            const float gy = (c * 256 + (rank >> 8)) * (2.0f / 1535.0f) - 1.0f;
            const long base = (((long)b * 1536 + over * 256 + qy) * 256 + qx) * 2;
            out2[base + 0] = gx;
            out2[base + 1] = gy;
            ++over;
            ++count;
        }
    }
    out4[(long)b * QTOT + q] = 1.0f / fmaxf((float)count, 1.0f);
}

// ---------------------------------------------------------------------------
extern "C" void kernel_launch(void* const* d_in, const int* in_sizes, int n_in,
                              void* d_out, int out_size, void* d_ws, size_t ws_size,
                              hipStream_t stream) {
    const float* ego = (const float*)d_in[0];
    float* out = (float*)d_out;

    const int Bn = in_sizes[0] / 96;             // (B,6,4,4) -> B
    const long n1 = (long)Bn * 6 * MAXLEN * 2;   // queries_rebatch_grid
    const long n2 = (long)Bn * 6 * 256 * 256 * 2;// restore_bev_grid
    const long n3 = (long)Bn * 6 * MAXLEN * 8;   // reference_points_rebatch

    float* out1 = out;
    float* out2 = out + n1;
    float* out3 = out + n1 + n2;
    float* out4 = out + n1 + n2 + n3;

    int* vis  = (int*)d_ws;
    int* incl = vis + (long)Bn * 6 * QTOT;

    const long fillN = n1 + n2 + n3;
    hipLaunchKernelGGL(fill_kernel, dim3((unsigned)((fillN + 255) / 256)), dim3(256), 0,
                       stream, out, n1 + n2, fillN);
    hipLaunchKernelGGL(proj_vis_wmma, dim3(QTOT / 32, 3 * Bn), dim3(256), 0,
                       stream, ego, vis);
    hipLaunchKernelGGL(scan_kernel, dim3(6 * Bn), dim3(1024), 0,
                       stream, vis, incl);
    hipLaunchKernelGGL(scatter_q, dim3(QTOT / 256, 6 * Bn), dim3(256), 0,
                       stream, ego, vis, incl, out1, out3);
    hipLaunchKernelGGL(restore_counts, dim3(QTOT / 256, Bn), dim3(256), 0,
                       stream, vis, incl, out2, out4);
}